// DecNP_21397527069039
// MI455X (gfx1250) — compile-verified
//
#include <hip/hip_runtime.h>

typedef __attribute__((ext_vector_type(2))) float v2f;
typedef __attribute__((ext_vector_type(8))) float v8f;

#define KNN_EPS 1e-8f
#define BATCH 4

// Branchless sorted top-3 insert (strict <; ascending scan order keeps
// earliest index on ties, matching top_k).
__device__ __forceinline__ void top3_insert(float dist, int si,
                                            float bd[3], int bi[3])
{
    const bool lt0 = dist < bd[0];
    const bool lt1 = dist < bd[1];
    const bool lt2 = dist < bd[2];
    const float nd2 = lt1 ? bd[1] : (lt2 ? dist : bd[2]);
    const int   ni2 = lt1 ? bi[1] : (lt2 ? si   : bi[2]);
    const float nd1 = lt0 ? bd[0] : (lt1 ? dist : bd[1]);
    const int   ni1 = lt0 ? bi[0] : (lt1 ? si   : bi[1]);
    bd[0] = lt0 ? dist : bd[0];
    bi[0] = lt0 ? si   : bi[0];
    bd[1] = nd1;  bi[1] = ni1;
    bd[2] = nd2;  bi[2] = ni2;
}

// Tie-aware variant for merging the two half-wave lists (prefer lower index
// on exact distance ties, emulating top_k appearance order).
__device__ __forceinline__ void top3_insert_tie(float dist, int si,
                                                float bd[3], int bi[3])
{
    const bool lt0 = (dist < bd[0]) || (dist == bd[0] && si < bi[0]);
    const bool lt1 = (dist < bd[1]) || (dist == bd[1] && si < bi[1]);
    const bool lt2 = (dist < bd[2]) || (dist == bd[2] && si < bi[2]);
    const float nd2 = lt1 ? bd[1] : (lt2 ? dist : bd[2]);
    const int   ni2 = lt1 ? bi[1] : (lt2 ? si   : bi[2]);
    const float nd1 = lt0 ? bd[0] : (lt1 ? dist : bd[1]);
    const int   ni1 = lt0 ? bi[0] : (lt1 ? si   : bi[1]);
    bd[0] = lt0 ? dist : bd[0];
    bi[0] = lt0 ? si   : bi[0];
    bd[1] = nd1;  bi[1] = ni1;
    bd[2] = nd2;  bi[2] = ni2;
}

// ---------------------------------------------------------------------------
// Kernel 1: 3-NN search via V_WMMA_F32_16X16X4_F32.
// One wave handles 16 target points (N dim).  Source points stream through M
// in chunks of 16, two chunks per loop iteration.
//   A row m = (-2sx, -2sy, -2sz, |s|^2),  B col n = (tx, ty, tz, 1)
//   D[m][n] = -2 s.t + |s|^2 ;  dist = D + |t|^2 (per-lane add)
// 32b 16x4 A layout: lane<16 -> (K0,K1) of M=lane ; lane>=16 -> (K2,K3) of
// M=lane-16.  B mirrors with N.  D: VGPR r -> M=r (lane<16) / M=8+r.
// ---------------------------------------------------------------------------
__global__ __launch_bounds__(32) void DecNP_knn_wmma_kernel(
    const float* __restrict__ tgt_xyz,  // (B, n1, 3)
    const float* __restrict__ src_xyz,  // (B, n2, 3)
    int n1, int n2,
    int*   __restrict__ idx_out,        // (B, n1, 3)
    float* __restrict__ w_out)          // (B, n1, 3)
{
    const int lane    = threadIdx.x;           // 0..31
    const int t_local = lane & 15;
    const bool hi     = lane >= 16;
    const int b       = blockIdx.y;
    const int t       = blockIdx.x * 16 + t_local;

    const float* tp = tgt_xyz + ((size_t)b * n1 + t) * 3;
    const float tx = tp[0], ty = tp[1], tz = tp[2];
    const float tn2 = tx * tx + ty * ty + tz * tz;

    v2f bvec;
    bvec.x = hi ? tz   : tx;   // K2 : K0
    bvec.y = hi ? 1.0f : ty;   // K3 : K1

    float bd[3] = {3.0e38f, 3.0e38f, 3.0e38f};
    int   bi[3] = {0, 0, 0};

    const float* src_b = src_xyz + (size_t)b * n2 * 3;
    const int moff = hi ? 8 : 0;

    // All n2 values (32,128,512,2048) are multiples of 32: process 2 chunks
    // of 16 source points per iteration -> 2 independent WMMAs in flight.
    for (int s0 = 0; s0 < n2; s0 += 32) {
        const float* sp0 = src_b + (size_t)(s0 + t_local) * 3;
        const float* sp1 = sp0 + 16 * 3;

        const float sx0 = sp0[0], sy0 = sp0[1], sz0 = sp0[2];
        const float sx1 = sp1[0], sy1 = sp1[1], sz1 = sp1[2];
        const float sn0 = sx0 * sx0 + sy0 * sy0 + sz0 * sz0;
        const float sn1 = sx1 * sx1 + sy1 * sy1 + sz1 * sz1;

        v2f a0, a1;
        a0.x = hi ? (-2.0f * sz0) : (-2.0f * sx0);
        a0.y = hi ? sn0           : (-2.0f * sy0);
        a1.x = hi ? (-2.0f * sz1) : (-2.0f * sx1);
        a1.y = hi ? sn1           : (-2.0f * sy1);

        v8f z0 = {}, z1 = {};
        v8f d0 = __builtin_amdgcn_wmma_f32_16x16x4_f32(
            false, a0, false, bvec, (short)0, z0, false, false);
        v8f d1 = __builtin_amdgcn_wmma_f32_16x16x4_f32(
            false, a1, false, bvec, (short)0, z1, false, false);

        const int mb0 = s0 + moff;
        const int mb1 = s0 + 16 + moff;
#pragma unroll
        for (int r = 0; r < 8; ++r)
            top3_insert(d0[r] + tn2, mb0 + r, bd, bi);
#pragma unroll
        for (int r = 0; r < 8; ++r)
            top3_insert(d1[r] + tn2, mb1 + r, bd, bi);
    }

    // Merge the two half-wave lists (lanes L and L+16 cover disjoint M rows
    // of the same target t): insert partner's sorted 3 with index tie-break.
#pragma unroll
    for (int j = 0; j < 3; ++j) {
        const float od = __shfl_xor(bd[j], 16, 32);
        const int   oi = __shfl_xor(bi[j], 16, 32);
        top3_insert_tie(od, oi, bd, bi);
    }

    if (lane < 16) {
        const float w0 = 1.0f / (bd[0] + KNN_EPS);
        const float w1 = 1.0f / (bd[1] + KNN_EPS);
        const float w2 = 1.0f / (bd[2] + KNN_EPS);
        const float wsum = w0 + w1 + w2;
        const size_t base = ((size_t)b * n1 + t) * 3;
        idx_out[base + 0] = bi[0];
        idx_out[base + 1] = bi[1];
        idx_out[base + 2] = bi[2];
        w_out[base + 0] = w0 / wsum;
        w_out[base + 1] = w1 / wsum;
        w_out[base + 2] = w2 / wsum;
    }
}

// ---------------------------------------------------------------------------
// Kernel 2: concat(points1, inverse-distance-interp(points2)) -> out.
// One thread per target point; coalesced writes along n; gathers hit L2
// (source feature map <= 63 MB << 192 MB global L2).
// ---------------------------------------------------------------------------
__global__ void DecNP_interp_kernel(
    const float* __restrict__ points1,  // (B, c1, n1)
    const float* __restrict__ points2,  // (B, c2, n2)
    const int*   __restrict__ idx,      // (B, n1, 3)
    const float* __restrict__ w,        // (B, n1, 3)
    float* __restrict__ out,            // (B, c1+c2, n1)
    int c1, int c2, int n1, int n2)
{
    const int n = blockIdx.x * blockDim.x + threadIdx.x;
    const int b = blockIdx.y;
    if (n >= n1) return;

    const size_t ibase = ((size_t)b * n1 + n) * 3;
    const int   i0 = idx[ibase + 0], i1 = idx[ibase + 1], i2 = idx[ibase + 2];
    const float w0 = w[ibase + 0],  w1 = w[ibase + 1],  w2 = w[ibase + 2];

    const int cout = c1 + c2;
    const float* p1b = points1 + (size_t)b * c1 * n1;
    const float* p2b = points2 + (size_t)b * c2 * n2;
    float*       ob  = out     + (size_t)b * cout * n1;

#pragma unroll 4
    for (int c = 0; c < c1; ++c)
        ob[(size_t)c * n1 + n] = p1b[(size_t)c * n1 + n];

#pragma unroll 4
    for (int c = 0; c < c2; ++c) {
        const float* f = p2b + (size_t)c * n2;
        ob[(size_t)(c1 + c) * n1 + n] =
            fmaf(w0, f[i0], fmaf(w1, f[i1], w2 * f[i2]));
    }
}

// ---------------------------------------------------------------------------
// Host-side stage driver.
// ---------------------------------------------------------------------------
static void run_stage(const float* tgt_xyz, const float* src_xyz,
                      const float* points1, const float* points2, float* out,
                      int n1, int n2, int c1, int c2,
                      int* idxBuf, float* wBuf, hipStream_t stream)
{
    dim3 g1(n1 / 16, BATCH), b1(32);
    DecNP_knn_wmma_kernel<<<g1, b1, 0, stream>>>(tgt_xyz, src_xyz, n1, n2,
                                                 idxBuf, wBuf);
    dim3 g2((n1 + 255) / 256, BATCH), b2(256);
    DecNP_interp_kernel<<<g2, b2, 0, stream>>>(points1, points2, idxBuf, wBuf,
                                               out, c1, c2, n1, n2);
}

extern "C" void kernel_launch(void* const* d_in, const int* in_sizes, int n_in,
                              void* d_out, int out_size, void* d_ws, size_t ws_size,
                              hipStream_t stream)
{
    // setup_inputs() dict order: xyz0,x0, xyz1,x1, xyz2,x2, xyz3,x3, xyz4,x4
    const float* xyz0 = (const float*)d_in[0];  // (4,8192,3)
    const float* x0   = (const float*)d_in[1];  // (4,  64,8192)
    const float* xyz1 = (const float*)d_in[2];  // (4,2048,3)
    const float* x1   = (const float*)d_in[3];  // (4, 128,2048)
    const float* xyz2 = (const float*)d_in[4];  // (4, 512,3)
    const float* x2   = (const float*)d_in[5];  // (4, 256, 512)
    const float* xyz3 = (const float*)d_in[6];  // (4, 128,3)
    const float* x3   = (const float*)d_in[7];  // (4, 512, 128)
    const float* xyz4 = (const float*)d_in[8];  // (4,  32,3)
    const float* x4   = (const float*)d_in[9];  // (4,1024,  32)

    // Workspace carve-out (floats).
    float* ws = (float*)d_ws;
    const size_t S0 = (size_t)BATCH * 1536 * 128;   // stage-0 out  (~3 MiB)
    const size_t S1 = (size_t)BATCH * 1792 * 512;   // stage-1 out  (~14 MiB)
    const size_t S2 = (size_t)BATCH * 1920 * 2048;  // stage-2 out  (~60 MiB)
    float* featA = ws;
    float* featB = featA + S0;
    float* featC = featB + S1;
    int*   idxBuf = (int*)(featC + S2);             // B*8192*3 ints
    float* wBuf   = (float*)(idxBuf + (size_t)BATCH * 8192 * 3);

    float* outF = (float*)d_out;                    // (4,1984,8192)

    // Stage 0: 32 -> 128
    run_stage(xyz3, xyz4, x3, x4,    featA, 128,  32,   512, 1024, idxBuf, wBuf, stream);
    // Stage 1: 128 -> 512
    run_stage(xyz2, xyz3, x2, featA, featB, 512,  128,  256, 1536, idxBuf, wBuf, stream);
    // Stage 2: 512 -> 2048
    run_stage(xyz1, xyz2, x1, featB, featC, 2048, 512,  128, 1792, idxBuf, wBuf, stream);
    // Stage 3: 2048 -> 8192 -> final output
    run_stage(xyz0, xyz1, x0, featC, outF,  8192, 2048, 64,  1920, idxBuf, wBuf, stream);
}